// CombinedLoss_5677946765322
// MI455X (gfx1250) — compile-verified
//
#include <hip/hip_runtime.h>
#include <math.h>

// ---------------------------------------------------------------------------
// CombinedLoss (dice + boundary(EDT) + focal) for (8,1,256,256) f32.
//
// Roofline: 4 MB of input @ 23.3 TB/s is ~170ns -> the whole thing is launch/
// latency bound; the only real compute is the exact distance transform
// (~268M fma+min ops brute force, microseconds of VALU). All DT values are
// small integers representable exactly in f32, so brute-force min-plus is
// bit-exact vs. the reference's O(n^2) formulation.
//
// WMMA use: final partial-sum reduction uses V_WMMA_F32_16X16X4_F32 with an
// all-ones A matrix => D = ones x B + C is an exact-f32 64-way adder with f32
// accumulation, independent of B's lane layout. (f16/fp8 WMMA would lose
// precision in the loss sums; the x4 f32 variant does not.)
// ---------------------------------------------------------------------------

typedef __attribute__((ext_vector_type(2))) float v2f;
typedef __attribute__((ext_vector_type(8))) float v8f;

#define Bn    8
#define Hn    256
#define Wn    256
#define HW    (Hn * Wn)          // 65536
#define NTOT  (Bn * HW)          // 524288
#define INF_F 131138.0f          // sum(n*n)+1 = 64+1+65536+65536+1
#define BIGF  3.0e38f
#define NBLK3 256                // blocks in kernel 3 (2048 rows / 8)

// ---------------- Kernel 1: mask + B-axis DT (n=8, C axis is trivial) ------
__global__ __launch_bounds__(256) void k_dt_b(const float* __restrict__ targets,
                                              float* __restrict__ f) {
  int pos = blockIdx.x * 256 + threadIdx.x;   // 0..HW-1, coalesced per b-slice
  float f0[Bn];
#pragma unroll
  for (int b = 0; b < Bn; ++b) {
    float t = targets[b * HW + pos];
    f0[b] = (t > 0.5f) ? INF_F : 0.0f;
  }
#pragma unroll
  for (int b = 0; b < Bn; ++b) {
    float m = BIGF;
#pragma unroll
    for (int bp = 0; bp < Bn; ++bp) {
      float d = (float)(b - bp);
      m = fminf(m, fmaf(d, d, f0[bp]));
    }
    f[b * HW + pos] = m;
  }
}

// ---------------- Kernel 2: H-axis DT, in place, LDS-tiled -----------------
// Block handles one (b, 32-wide w tile): 256x32 f32 tile = 32KB LDS.
__global__ __launch_bounds__(256) void k_dt_h(float* __restrict__ f) {
  __shared__ float tile[Hn][32];
  int b    = blockIdx.x >> 3;
  int wt   = blockIdx.x & 7;
  int base = b * HW + wt * 32;
  int tid  = threadIdx.x;

  for (int idx = tid; idx < Hn * 32; idx += 256) {          // coalesced load
    int h = idx >> 5, w = idx & 31;
    tile[h][w] = f[base + h * Wn + w];
  }
  __syncthreads();

  int w  = tid & 31;          // lane -> w   (conflict-free LDS columns)
  int ig = (tid >> 5) * 32;   // wave -> 32 consecutive output rows i
  for (int r = 0; r < 32; ++r) {
    int i = ig + r;
    float m = BIGF;
    for (int j = 0; j < Hn; ++j) {
      float d = (float)(i - j);
      m = fminf(m, fmaf(d, d, tile[j][w]));
    }
    f[base + i * Wn + w] = m;  // in place: tile already fully staged in LDS
  }
}

// ---- Kernel 3: W-axis DT fused with sigmoid/dice/focal/boundary terms -----
// Block handles 8 contiguous rows (8*256 elements); deterministic fixed-order
// block reduction (no float atomics).
__global__ __launch_bounds__(256) void k_dt_w_loss(const float* __restrict__ f2,
                                                   const float* __restrict__ logits,
                                                   const float* __restrict__ targets,
                                                   float* __restrict__ partials) {
  __shared__ float rows[8][Wn];
  __shared__ float red[8][5];
  int tid   = threadIdx.x;
  int blk   = blockIdx.x;
  int gbase = blk * 8 * Wn;

  __builtin_prefetch(&logits[gbase + tid], 0, 0);   // global_prefetch_b8
  __builtin_prefetch(&targets[gbase + tid], 0, 0);

  for (int idx = tid; idx < 8 * Wn; idx += 256)     // contiguous, coalesced
    rows[idx >> 8][idx & 255] = f2[gbase + idx];
  __syncthreads();

  float s_p = 0.f, s_t = 0.f, s_pt = 0.f, s_b = 0.f, s_f = 0.f;
  for (int s = 0; s < 8; ++s) {
    float m = BIGF;
    for (int j = 0; j < Wn; ++j) {                  // LDS broadcast reads
      float d = (float)(tid - j);
      m = fminf(m, fmaf(d, d, rows[s][j]));
    }
    float dm = sqrtf(m);

    int   g  = gbase + s * Wn + tid;
    float L  = logits[g];
    float t  = targets[g];
    float p  = 1.0f / (1.0f + expf(-L));
    float om = 1.0f - p;
    s_p  += p;
    s_t  += t;
    s_pt += p * t;
    s_b  += dm * om * om;
    float ce = fmaxf(L, 0.0f) - L * t + log1pf(expf(-fabsf(L)));
    float pt = p * t + om * (1.0f - t);
    float u  = 1.0f - pt;
    s_f += 0.25f * u * u * ce;                      // ALPHA=0.25, GAMMA=2
  }

  float v[5] = {s_p, s_t, s_pt, s_b, s_f};
#pragma unroll
  for (int q = 0; q < 5; ++q)
#pragma unroll
    for (int off = 16; off > 0; off >>= 1)
      v[q] += __shfl_down(v[q], off);               // wave32 reduce
  if ((tid & 31) == 0) {
#pragma unroll
    for (int q = 0; q < 5; ++q) red[tid >> 5][q] = v[q];
  }
  __syncthreads();
  if (tid == 0) {
#pragma unroll
    for (int q = 0; q < 5; ++q) {
      float s = 0.f;
      for (int wv = 0; wv < 8; ++wv) s += red[wv][q];
      partials[q * NBLK3 + blk] = s;
    }
  }
}

// ---- Kernel 4: WMMA (f32 16x16x4) exact reduction of 5x256 partials -------
// A = all-ones -> D[m][n] = sum_k B[k][n] + C[m][n]; since every B register
// slot maps to a unique (k,n), sum_n D[0][n] totals all 64 staged values
// regardless of B's exact lane layout. Chained as accumulator (C=prev D).
__global__ __launch_bounds__(32) void k_final(const float* __restrict__ partials,
                                              float* __restrict__ out) {
  int lane = threadIdx.x;
  v2f ones;
  ones.x = 1.0f;
  ones.y = 1.0f;

  float sums[5];
#pragma unroll
  for (int q = 0; q < 5; ++q) {
    v8f acc = {};
#pragma unroll
    for (int c = 0; c < 4; ++c) {
      v2f bv;
      bv.x = partials[q * NBLK3 + c * 64 + lane];
      bv.y = partials[q * NBLK3 + c * 64 + 32 + lane];
      // 8 args: (neg_a, A, neg_b, B, c_mod, C, reuse_a, reuse_b)
      acc = __builtin_amdgcn_wmma_f32_16x16x4_f32(
          false, ones, false, bv, (short)0, acc, false, false);
    }
    // acc[0]: lanes 0..15 hold D[0][0..15] (all D rows identical, A=ones)
    float s = acc[0];
#pragma unroll
    for (int off = 8; off > 0; off >>= 1)
      s += __shfl_down(s, off);
    sums[q] = s;  // valid on lane 0
  }

  if (lane == 0) {
    const float EPSf = 1e-6f;
    float dice = 1.0f - (2.0f * sums[2] + EPSf) / (sums[0] + sums[1] + EPSf);
    float invN = 1.0f / (float)NTOT;
    out[0] = 1.0f * dice + 0.5f * (sums[3] * invN) + 1.0f * (sums[4] * invN);
  }
}

// ---------------------------------------------------------------------------
extern "C" void kernel_launch(void* const* d_in, const int* in_sizes, int n_in,
                              void* d_out, int out_size, void* d_ws, size_t ws_size,
                              hipStream_t stream) {
  (void)in_sizes; (void)n_in; (void)out_size; (void)ws_size;
  const float* logits  = (const float*)d_in[0];
  const float* targets = (const float*)d_in[1];
  float* f        = (float*)d_ws;        // NTOT floats (2 MB), reused in place
  float* partials = f + NTOT;            // 5 * NBLK3 floats
  float* out      = (float*)d_out;

  k_dt_b    <<<HW / 256, 256, 0, stream>>>(targets, f);
  k_dt_h    <<<64,       256, 0, stream>>>(f);
  k_dt_w_loss<<<NBLK3,   256, 0, stream>>>(f, logits, targets, partials);
  k_final   <<<1,         32, 0, stream>>>(partials, out);
}